// IdentitySlotUNet_36043365548302
// MI455X (gfx1250) — compile-verified
//
#include <hip/hip_runtime.h>
#include <math.h>

// ---------------------------------------------------------------------------
// IdentitySlotUNet fused injection for MI455X (gfx1250, wave32, WMMA + TDM).
//
// Roofline: ~6 GFLOP vs ~754 MB HBM traffic -> memory bound. Design goal:
// read hidden_states exactly once (TDM DMA into LDS tiles), write output once.
// Matrix ops use V_WMMA_F32_16X16X4_F32 (N=16 identities == WMMA tile dim).
// ---------------------------------------------------------------------------

typedef __attribute__((ext_vector_type(2))) float v2f;
typedef __attribute__((ext_vector_type(8))) float v8f;
typedef __attribute__((ext_vector_type(4))) unsigned int v4u;
typedef __attribute__((ext_vector_type(8))) int v8i_;
typedef __attribute__((ext_vector_type(4))) int v4i_;

#define BB 8
#define CC 1280
#define HH 96
#define WW 96
#define HWP (HH * WW)     // 9216; W=96 divisible by 16 so a 16-pixel tile
                          // never crosses an image row.
#define NN 16
#define DD 512
#define STRENGTH 1.5f
#define INV_T (1.0f / (0.5f + 1e-6f))
#define EPSV 1e-6f

#if defined(__has_builtin)
#if __has_builtin(__builtin_amdgcn_tensor_load_to_lds) && \
    __has_builtin(__builtin_amdgcn_s_wait_tensorcnt)
#define USE_TDM 1
#endif
#endif

// D = A(16x4) * B(4x16) + C, fp32 accumulate.
// A frag: lane m=l&15; VGPR0 holds K = 2*(l>>4), VGPR1 holds K+1.
// B frag: lane n=l&15; VGPR0 holds K = 2*(l>>4), VGPR1 holds K+1.
// C/D:    lane col=l&15; VGPR g holds row g + 8*(l>>4).
__device__ inline v8f wmma4(v2f a, v2f b, v8f c) {
  return __builtin_amdgcn_wmma_f32_16x16x4_f32(
      /*neg_a=*/false, a, /*neg_b=*/false, b,
      /*c_mod=*/(short)0, c, /*reuse_a=*/false, /*reuse_b=*/false);
}

__device__ inline float gelu_exact(float x) {
  return 0.5f * x * (1.0f + erff(x * 0.70710678118654752f));
}

// ---------------- MLP kernel 1: h1 = gelu(E @ W1 + b1)  [128 x 1280] -------
__global__ __launch_bounds__(32) void mlp_gemm1(const float* __restrict__ emb,
                                                const float* __restrict__ w1,
                                                const float* __restrict__ pb1,
                                                float* __restrict__ h1) {
  const int l = threadIdx.x & 31, hi = l >> 4, m = l & 15;
  const int rt = blockIdx.x / (CC / 16);
  const int ct = blockIdx.x % (CC / 16);
  const int r0 = rt * 16, c0 = ct * 16;
  v8f acc = {};
  for (int k0 = 0; k0 < DD; k0 += 4) {
    const int k = k0 + 2 * hi;
    v2f a, b;
    a.x = emb[(r0 + m) * DD + k];
    a.y = emb[(r0 + m) * DD + k + 1];
    b.x = w1[k * CC + c0 + m];
    b.y = w1[(k + 1) * CC + c0 + m];
    acc = wmma4(a, b, acc);
  }
  for (int g = 0; g < 8; ++g) {
    const int row = r0 + g + 8 * hi, col = c0 + m;
    h1[row * CC + col] = gelu_exact(acc[g] + pb1[col]);
  }
}

// ---------------- MLP kernel 2: proj_raw = h1 @ W2 + b2  [128 x 1280] ------
__global__ __launch_bounds__(32) void mlp_gemm2(const float* __restrict__ h1,
                                                const float* __restrict__ w2,
                                                const float* __restrict__ pb2,
                                                float* __restrict__ proj) {
  const int l = threadIdx.x & 31, hi = l >> 4, m = l & 15;
  const int rt = blockIdx.x / (CC / 16);
  const int ct = blockIdx.x % (CC / 16);
  const int r0 = rt * 16, c0 = ct * 16;
  v8f acc = {};
  for (int k0 = 0; k0 < CC; k0 += 4) {
    const int k = k0 + 2 * hi;
    v2f a, b;
    a.x = h1[(r0 + m) * CC + k];
    a.y = h1[(r0 + m) * CC + k + 1];
    b.x = w2[k * CC + c0 + m];
    b.y = w2[(k + 1) * CC + c0 + m];
    acc = wmma4(a, b, acc);
  }
  for (int g = 0; g < 8; ++g) {
    const int row = r0 + g + 8 * hi, col = c0 + m;
    proj[row * CC + col] = acc[g] + pb2[col];
  }
}

// ---------------- MLP kernel 3: row L2-normalize * sqrt(C), in place -------
__global__ __launch_bounds__(256) void proj_normalize(float* __restrict__ proj) {
  __shared__ float red[256];
  const int r = blockIdx.x;       // 0..127
  const int tid = threadIdx.x;
  float s = 0.f;
  for (int c = tid; c < CC; c += 256) {
    const float v = proj[r * CC + c];
    s += v * v;
  }
  red[tid] = s;
  __syncthreads();
  for (int off = 128; off > 0; off >>= 1) {
    if (tid < off) red[tid] += red[tid + off];
    __syncthreads();
  }
  const float scale = sqrtf((float)CC) / (sqrtf(red[0]) + EPSV);
  for (int c = tid; c < CC; c += 256) proj[r * CC + c] *= scale;
}

// ---------------- Fused: normalize / scores / weights / inject / add -------
// Grid: B * (HW/16) blocks, 256 threads (8 wave32). One block owns a
// 16-pixel tile across all 1280 channels; the full [1280 x 16] hidden slab
// is DMA'd into LDS by the Tensor Data Mover so HBM sees exactly one read +
// one write per element and no VGPR staging for the fill.
__global__ __launch_bounds__(256) void fused_inject(
    const float* __restrict__ hid, const float* __restrict__ proj,
    const float* __restrict__ bbox, float* __restrict__ out) {
  extern __shared__ float smem[];
  float* s_hid = smem;           // [1280][16]  = 20480 f  (at LDS offset 0!)
  float* s_sc = smem + 20480;    // [16n][16p]  = 256 f  (scores accum)
  float* s_w = smem + 20736;     // [16n][16p]  = 256 f  (weights)
  float* s_part = smem + 20992;  // [16g][16p]  = 256 f  (sumsq partials)
  float* s_inv = smem + 21248;   // [16p]       1/(||h||+eps)
  float* s_den = smem + 21264;   // [16p]

  const int tid = threadIdx.x;
  const int lane = tid & 31;
  const int wv = tid >> 5;       // wave 0..7
  const int hi = lane >> 4;
  const int m = lane & 15;

  const int b = blockIdx.x / (HWP / 16);
  const int tile = blockIdx.x % (HWP / 16);
  const int p0 = tile * 16;

  // ---- stage 0: hidden tile -> LDS ----
  const float* hb = hid + (size_t)b * CC * HWP;
#ifdef USE_TDM
  // TDM 2D tile: X = 16 pixels (contiguous), Y = 1280 channels with row
  // stride HWP elements. TDM packs rows back-to-back in LDS, which is
  // exactly the s_hid[c*16 + i] layout. One DMA per block, wave 0 issues
  // (EXEC is ignored by tensor ops; one instruction = one transfer).
  if (wv == 0) {
    const unsigned long long ga =
        (unsigned long long)(const void*)(hb + (size_t)p0);
    v4u g0;
    g0.x = 1u;                                    // count=1, user descriptor
    g0.y = 0u;                                    // lds_addr: dynamic LDS base
    g0.z = (unsigned int)(ga & 0xffffffffull);    // global_addr[31:0]
    g0.w = (unsigned int)((ga >> 32) & 0x01ffffffull) | (2u << 30);  // type=2
    v8i_ g1;
    g1[0] = (int)(2u << 16);       // workgroup_mask=0, data_size=2 (4 bytes)
    g1[1] = (int)(16u << 16);      // tensor_dim0[15:0] = 16
    g1[2] = (int)(1280u << 16);    // tensor_dim0[31:16]=0, tensor_dim1=1280
    g1[3] = (int)(16u << 16);      // tensor_dim1[31:16]=0, tile_dim0=16
    g1[4] = (int)1280;             // tile_dim1=1280, tile_dim2=0 (unused)
    g1[5] = (int)HWP;              // tensor_dim0_stride[31:0] = 9216
    g1[6] = 0;                     // stride0[47:32]=0, stride1[15:0]=0
    g1[7] = 0;
    v4i_ g2 = (v4i_)0, g3 = (v4i_)0;  // 2D tensor: groups 2/3 unused
#if __has_include(<hip/amd_detail/amd_gfx1250_TDM.h>)
    v8i_ g4 = (v8i_)0;
    __builtin_amdgcn_tensor_load_to_lds(g0, g1, g2, g3, g4, 0);
#else
    __builtin_amdgcn_tensor_load_to_lds(g0, g1, g2, g3, 0);
#endif
    __builtin_amdgcn_s_wait_tensorcnt(0);
  }
#else
  // Fallback: float4 cooperative copy through VGPRs.
  {
    float4* s_hid4 = (float4*)s_hid;
    for (int it = 0; it < 20; ++it) {
      const int idx = tid + it * 256;  // float4 index 0..5119
      const int c = idx >> 2;
      const int q = idx & 3;
      s_hid4[idx] = *(const float4*)(hb + (size_t)c * HWP + p0 + q * 4);
    }
  }
#endif
  s_sc[tid] = 0.f;
  __syncthreads();

  // ---- stage 1: per-pixel channel L2 norm ----
  {
    const int i = tid & 15, g = tid >> 4;
    float s = 0.f;
    for (int c = g; c < CC; c += 16) {
      const float v = s_hid[c * 16 + i];
      s += v * v;
    }
    s_part[g * 16 + i] = s;
  }
  __syncthreads();
  if (tid < 16) {
    float s = 0.f;
    for (int g = 0; g < 16; ++g) s += s_part[g * 16 + tid];
    s_inv[tid] = 1.0f / (sqrtf(s) + EPSV);
  }
  __syncthreads();

  // ---- stage 2: scores[n,p] = sum_c proj[n,c]*hidn[c,p] (WMMA, K split) ---
  const float* pjb = proj + (size_t)b * NN * CC;
  {
    v8f acc = {};
    const float inv = s_inv[m];
    const int kbase = wv * (CC / 8);  // 160 channels per wave
    for (int s = 0; s < (CC / 8) / 4; ++s) {
      const int k = kbase + 4 * s + 2 * hi;
      v2f a, bf;
      a.x = pjb[m * CC + k];
      a.y = pjb[m * CC + k + 1];
      bf.x = s_hid[k * 16 + m] * inv;
      bf.y = s_hid[(k + 1) * 16 + m] * inv;
      acc = wmma4(a, bf, acc);
    }
    for (int g = 0; g < 8; ++g)
      atomicAdd(&s_sc[(g + 8 * hi) * 16 + m], acc[g]);  // ds_add_f32
  }
  __syncthreads();

  // ---- stage 3: masked sigmoid weights, normalize over n, pow 1.2 ----
  {
    const int i = tid & 15, n = tid >> 4;
    const int pg = p0 + i;
    const int y = pg / WW, x = pg % WW;
    bool any_box = false;
    for (int j = 0; j < NN; ++j) {
      const int ax1 = (int)floorf(fminf(fmaxf(bbox[j * 4 + 0] * WW, 0.f), (float)WW));
      const int ay1 = (int)floorf(fminf(fmaxf(bbox[j * 4 + 1] * HH, 0.f), (float)HH));
      const int ax2 = (int)floorf(fminf(fmaxf(bbox[j * 4 + 2] * WW, 0.f), (float)WW));
      const int ay2 = (int)floorf(fminf(fmaxf(bbox[j * 4 + 3] * HH, 0.f), (float)HH));
      if (ax2 > ax1 && ay2 > ay1) any_box = true;
    }
    const int x1 = (int)floorf(fminf(fmaxf(bbox[n * 4 + 0] * WW, 0.f), (float)WW));
    const int y1 = (int)floorf(fminf(fmaxf(bbox[n * 4 + 1] * HH, 0.f), (float)HH));
    const int x2 = (int)floorf(fminf(fmaxf(bbox[n * 4 + 2] * WW, 0.f), (float)WW));
    const int y2 = (int)floorf(fminf(fmaxf(bbox[n * 4 + 3] * HH, 0.f), (float)HH));
    const bool in_box = (y >= y1 && y < y2 && x >= x1 && x < x2);
    const float mask = (!any_box || in_box) ? 1.f : 0.f;
    const float sc = s_sc[tid];
    s_w[tid] = (mask != 0.f) ? (1.0f / (1.0f + expf(-sc * INV_T))) : 0.f;
  }
  __syncthreads();
  if (tid < 16) {
    float s = 0.f;
    for (int n = 0; n < NN; ++n) s += s_w[n * 16 + tid];
    s_den[tid] = s + EPSV;
  }
  __syncthreads();
  {
    const int i = tid & 15;
    s_w[tid] = powf(s_w[tid] / s_den[i], 1.2f);
  }
  __syncthreads();

  // ---- stage 4: out[c,p] = hid[c,p] + 1.5 * sum_n w[n,p]*proj[n,c] -------
  // A = proj^T[16c x 16n], B = w[16n x 16p], K=16 (4 WMMA steps).
  float* ob = out + (size_t)b * CC * HWP;
  for (int cg = wv; cg < CC / 16; cg += 8) {  // 10 channel-groups per wave
    v8f acc = {};
    for (int k0 = 0; k0 < 16; k0 += 4) {
      const int k = k0 + 2 * hi;
      v2f a, bf;
      a.x = pjb[k * CC + cg * 16 + m];
      a.y = pjb[(k + 1) * CC + cg * 16 + m];
      bf.x = s_w[k * 16 + m];
      bf.y = s_w[(k + 1) * 16 + m];
      acc = wmma4(a, bf, acc);
    }
    for (int g = 0; g < 8; ++g) {
      const int c = cg * 16 + g + 8 * hi;
      ob[(size_t)c * HWP + p0 + m] = s_hid[c * 16 + m] + STRENGTH * acc[g];
    }
  }
}

// ---------------------------------------------------------------------------
extern "C" void kernel_launch(void* const* d_in, const int* in_sizes, int n_in,
                              void* d_out, int out_size, void* d_ws,
                              size_t ws_size, hipStream_t stream) {
  const float* hid = (const float*)d_in[0];   // [8,1280,96,96]
  const float* emb = (const float*)d_in[1];   // [8,16,512]
  const float* bbox = (const float*)d_in[2];  // [16,4]
  const float* w1 = (const float*)d_in[3];    // [512,1280]
  const float* pb1 = (const float*)d_in[4];   // [1280]
  const float* w2 = (const float*)d_in[5];    // [1280,1280]
  const float* pb2 = (const float*)d_in[6];   // [1280]
  float* out = (float*)d_out;

  float* h1 = (float*)d_ws;                   // [128,1280]
  float* proj = h1 + (size_t)BB * NN * CC;    // [128,1280]

  const int gemm_blocks = (BB * NN / 16) * (CC / 16);  // 640
  mlp_gemm1<<<gemm_blocks, 32, 0, stream>>>(emb, w1, pb1, h1);
  mlp_gemm2<<<gemm_blocks, 32, 0, stream>>>(h1, w2, pb2, proj);
  proj_normalize<<<BB * NN, 256, 0, stream>>>(proj);

  const size_t smem_bytes = (size_t)21280 * sizeof(float);  // ~85 KB of 320 KB/WGP
  (void)hipFuncSetAttribute((const void*)fused_inject,
                            hipFuncAttributeMaxDynamicSharedMemorySize,
                            (int)smem_bytes);
  fused_inject<<<BB * (HWP / 16), 256, smem_bytes, stream>>>(hid, proj, bbox, out);
}